// SSDLoss_88682484728390
// MI455X (gfx1250) — compile-verified
//
#include <hip/hip_runtime.h>
#include <hip/hip_bf16.h>

#define BATCH   32
#define PRIORS  8732
#define NCLS    81
#define TILE_P  128
#define TPB1    128   // pass1: one thread per prior, 4 waves
#define TPB2    256   // pass2: selection/scan kernel

typedef float v2f __attribute__((ext_vector_type(2)));
typedef float v8f __attribute__((ext_vector_type(8)));

struct WsHeader {
    double sl1_sum;       // offset 0
    double posce_sum;     // offset 8
    double negce_sum;     // offset 16
    double pad0;          // offset 24
    int    num_pos[BATCH];// offset 32..159
    int    pad1[24];      // pad to 256 bytes
};

// Monotone map: larger float -> larger unsigned key.
__device__ __forceinline__ unsigned mono_key(float f) {
    unsigned u = __float_as_uint(f);
    return (u & 0x80000000u) ? ~u : (u | 0x80000000u);
}

// Exact f32 sum across the 32 lanes of a wave using two chained
// V_WMMA_F32_16X16X4_F32 ops with B = all-ones.
//   WMMA1: D1[m][n] = rowsum(A[m]) -> 16 row sums, each replicated over lanes.
//   Fold D1's 8 accumulators per lane into a new A; WMMA2 then produces the
//   total sum of all 64 original A entries in every element of D2.
// Lanes put (x, 0) into A, so the result is the wave-wide sum of x.
// EXEC must be all ones: call only from converged code.
__device__ __forceinline__ float wave_sum_wmma(float x) {
    v2f a;    a.x = x;   a.y = 0.0f;
    v2f ones; ones.x = 1.0f; ones.y = 1.0f;
    v8f c = {};
    c = __builtin_amdgcn_wmma_f32_16x16x4_f32(false, a, false, ones,
                                              (short)0, c, false, false);
    v2f a2;
    a2.x = c[0] + c[1] + c[2] + c[3];
    a2.y = c[4] + c[5] + c[6] + c[7];
    v8f d = {};
    d = __builtin_amdgcn_wmma_f32_16x16x4_f32(false, a2, false, ones,
                                              (short)0, d, false, false);
    return d[0];
}

__global__ void ssd_init(WsHeader* hdr) {
    int t = threadIdx.x;
    if (t == 0) {
        hdr->sl1_sum = 0.0;
        hdr->posce_sum = 0.0;
        hdr->negce_sum = 0.0;
    }
    if (t < BATCH) hdr->num_pos[t] = 0;
}

// Pass 1: stream confidence tile -> LDS via async-to-LDS, compute per-prior
// logsumexp / CE / background key; accumulate positive-side sums.
// 128 threads == 128 priors: every thread copies AND computes.
__global__ __launch_bounds__(TPB1) void ssd_pass1(
    const float* __restrict__ conf, const float* __restrict__ pred,
    const int* __restrict__ labels, const float* __restrict__ gt,
    WsHeader* hdr, unsigned* __restrict__ keys, float* __restrict__ negce)
{
    __shared__ float smem[TILE_P * NCLS];        // 41472 B
    __shared__ float wred[TPB1 / 32][4];

    const int tiles = (PRIORS + TILE_P - 1) / TILE_P;   // 69
    const int b    = blockIdx.x / tiles;
    const int tile = blockIdx.x % tiles;
    const int p0   = tile * TILE_P;
    const int nP   = min(TILE_P, PRIORS - p0);          // 128 or 28 (both %4==0)
    const int tid  = threadIdx.x;

    // --- async DMA: conf[b, p0:p0+nP, :] is contiguous; copy as float4s ---
    // ~20 b128 async issues per lane (ASYNCcnt max 63 outstanding per wave).
    const char* gbase =
        (const char*)(conf + ((size_t)b * PRIORS + p0) * NCLS);
    const int nf4 = (nP * NCLS) >> 2;  // exact: nP*81 divisible by 4
    unsigned lbase = (unsigned)(size_t)(&smem[0]); // low 32 bits = LDS offset
    for (int i = tid; i < nf4; i += TPB1) {
        unsigned loff = lbase + (unsigned)(i * 16);
        unsigned long long ga =
            (unsigned long long)gbase + (unsigned long long)i * 16ull;
        asm volatile("global_load_async_to_lds_b128 %0, %1, off"
                     :: "v"(loff), "v"(ga) : "memory");
    }
    asm volatile("s_wait_asynccnt 0x0" ::: "memory");
    __syncthreads();

    // --- per-prior math (row stride 81 words => conflict-free banks) ---
    float sl1 = 0.0f, posce = 0.0f, posf = 0.0f;
    if (tid < nP) {
        const int p = p0 + tid;
        const float* row = &smem[tid * NCLS];
        float m = row[0];
        #pragma unroll 4
        for (int j = 1; j < NCLS; ++j) m = fmaxf(m, row[j]);
        float s = 0.0f;
        #pragma unroll 4
        for (int j = 0; j < NCLS; ++j) s += __expf(row[j] - m);
        float lse = m + __logf(s);

        const size_t gidx = (size_t)b * PRIORS + p;
        const int lab = labels[gidx];
        const float ce = lse - row[lab];
        const float bg = lse - row[0];         // -log_softmax[..., 0]
        const bool pos = lab > 0;

        keys[gidx]  = pos ? 0u : mono_key(bg); // pos -> minimal key (== -inf)
        negce[gidx] = pos ? 0.0f : ce;

        if (pos) {
            posf = 1.0f;
            posce = ce;
            const float4 dp = ((const float4*)pred)[gidx];
            const float4 dg = ((const float4*)gt)[gidx];
            float dx[4] = {dp.x - dg.x, dp.y - dg.y, dp.z - dg.z, dp.w - dg.w};
            #pragma unroll
            for (int j = 0; j < 4; ++j) {
                float ad = fabsf(dx[j]);
                sl1 += (ad < 1.0f) ? 0.5f * dx[j] * dx[j] : ad - 0.5f;
            }
        }
    }

    // --- converged wave reductions on the matrix pipe ---
    float wsl1 = wave_sum_wmma(sl1);
    float wpce = wave_sum_wmma(posce);
    float wpos = wave_sum_wmma(posf);
    const int wid = tid >> 5, lane = tid & 31;
    if (lane == 0) {
        wred[wid][0] = wsl1; wred[wid][1] = wpce; wred[wid][2] = wpos;
    }
    __syncthreads();
    if (tid == 0) {
        float a = 0.0f, pc = 0.0f, np = 0.0f;
        for (int w = 0; w < TPB1 / 32; ++w) {
            a += wred[w][0]; pc += wred[w][1]; np += wred[w][2];
        }
        atomicAdd(&hdr->sl1_sum,   (double)a);
        atomicAdd(&hdr->posce_sum, (double)pc);
        atomicAdd(&hdr->num_pos[b], (int)(np + 0.5f));
    }
}

__device__ __forceinline__ int block_count_ge(const unsigned* ks, unsigned T,
                                              int* red, int tid) {
    int c = 0;
    for (int i = tid; i < PRIORS; i += TPB2) c += (ks[i] >= T) ? 1 : 0;
    red[tid] = c;
    __syncthreads();
    for (int s = TPB2 / 2; s > 0; s >>= 1) {
        if (tid < s) red[tid] += red[tid + s];
        __syncthreads();
    }
    int r = red[0];
    __syncthreads();
    return r;
}

// Pass 2: per batch, exact top-k (k = min(3*num_pos, P)) by key with stable
// (index-order) tie handling; sum negative CE over the selected set.
__global__ __launch_bounds__(TPB2) void ssd_pass2(
    WsHeader* hdr, const unsigned* __restrict__ keys,
    const float* __restrict__ negce)
{
    __shared__ unsigned ks[PRIORS];     // ~34.9 KB
    __shared__ int   red[TPB2];
    __shared__ float redf[TPB2];
    __shared__ int   scan_s[TPB2];

    const int b = blockIdx.x, tid = threadIdx.x;
    for (int i = tid; i < PRIORS; i += TPB2)
        ks[i] = keys[(size_t)b * PRIORS + i];
    __syncthreads();

    const int npos = hdr->num_pos[b];
    long long kk = (long long)npos * 3;
    const int k = (kk > PRIORS) ? PRIORS : (int)kk;
    if (k <= 0) return;  // uniform exit

    // Bitwise binary search: largest T with count(key >= T) >= k, i.e. the
    // k-th largest key value.
    unsigned T = 0u;
    for (int bit = 31; bit >= 0; --bit) {
        unsigned cand = T | (1u << bit);
        int c = block_count_ge(ks, cand, red, tid);
        if (c >= k) T = cand;
    }
    int cnt_gt = (T == 0xFFFFFFFFu) ? 0 : block_count_ge(ks, T + 1u, red, tid);
    const int needed = k - cnt_gt;   // ties to take, in ascending index order

    float sum = 0.0f;
    for (int i = tid; i < PRIORS; i += TPB2)
        if (ks[i] > T) sum += negce[(size_t)b * PRIORS + i];

    // Stable selection among equals: contiguous chunks + serial scan.
    const int C  = (PRIORS + TPB2 - 1) / TPB2;   // 35
    const int st = tid * C;
    const int en = min(st + C, PRIORS);
    int myeq = 0;
    for (int i = st; i < en; ++i) myeq += (ks[i] == T) ? 1 : 0;
    red[tid] = myeq;
    __syncthreads();
    if (tid == 0) {
        int run = 0;
        for (int j = 0; j < TPB2; ++j) { scan_s[j] = run; run += red[j]; }
    }
    __syncthreads();
    int rank = scan_s[tid];
    for (int i = st; i < en; ++i) {
        if (ks[i] == T) {
            if (rank < needed) sum += negce[(size_t)b * PRIORS + i];
            ++rank;
        }
    }

    redf[tid] = sum;
    __syncthreads();
    for (int s = TPB2 / 2; s > 0; s >>= 1) {
        if (tid < s) redf[tid] += redf[tid + s];
        __syncthreads();
    }
    if (tid == 0) atomicAdd(&hdr->negce_sum, (double)redf[0]);
}

__global__ void ssd_final(const WsHeader* hdr, float* out) {
    if (threadIdx.x == 0 && blockIdx.x == 0) {
        int np = 0;
        for (int i = 0; i < BATCH; ++i) np += hdr->num_pos[i];
        double dn = (double)np;
        out[0] = (float)(hdr->sl1_sum / dn);
        out[1] = (float)((hdr->posce_sum + hdr->negce_sum) / dn);
    }
}

extern "C" void kernel_launch(void* const* d_in, const int* in_sizes, int n_in,
                              void* d_out, int out_size, void* d_ws,
                              size_t ws_size, hipStream_t stream) {
    const float* conf   = (const float*)d_in[0];
    const float* pred   = (const float*)d_in[1];
    const int*   labels = (const int*)d_in[2];
    const float* gt     = (const float*)d_in[3];
    float* out = (float*)d_out;

    WsHeader* hdr = (WsHeader*)d_ws;
    unsigned* keys = (unsigned*)((char*)d_ws + 256);
    float* negce =
        (float*)((char*)d_ws + 256 + sizeof(unsigned) * (size_t)BATCH * PRIORS);

    ssd_init<<<1, 64, 0, stream>>>(hdr);
    const int tiles = (PRIORS + TILE_P - 1) / TILE_P;   // 69
    ssd_pass1<<<BATCH * tiles, TPB1, 0, stream>>>(conf, pred, labels, gt,
                                                  hdr, keys, negce);
    ssd_pass2<<<BATCH, TPB2, 0, stream>>>(hdr, keys, negce);
    ssd_final<<<1, 1, 0, stream>>>(hdr, out);
}